// QKNorm_317827580175
// MI455X (gfx1250) — compile-verified
//
#include <hip/hip_runtime.h>
#include <hip/hip_bf16.h>

typedef float f32x2 __attribute__((ext_vector_type(2)));
typedef float f32x8 __attribute__((ext_vector_type(8)));

// D = A(16x4 f32) * B(4x16 f32) + C(16x16 f32), wave32 WMMA.
__device__ __forceinline__ f32x8 wmma_f32(f32x2 a, f32x2 b, f32x8 c) {
  return __builtin_amdgcn_wmma_f32_16x16x4_f32(
      /*neg_a=*/false, a, /*neg_b=*/false, b,
      /*c_mod=*/(short)0, c, /*reuse_a=*/false, /*reuse_b=*/false);
}

#define NDIM 4096
#define DDIM 128
#define MDIM 16
#define PDIM 8192
#define HDIM 32
#define NTILES 513       // 8192/16 cache tiles + 1 new-rows tile
#define SPLITS 4
#define TPS 129          // tiles per split (ceil(513/4))
#define PARTF 2080       // floats per partial: 16 max + 16 sum + 16*128 out

// ---------------- Kernel 1: Q/K/V = X @ W{q,k,v} -------------------------
// One wave per 16x16 output tile. 768 tiles total (3 matrices x 256).
__global__ __launch_bounds__(256) void qkv_kernel(
    const float* __restrict__ X, const float* __restrict__ Wq,
    const float* __restrict__ Wk, const float* __restrict__ Wv,
    float* __restrict__ qkv) {
  const int lane = threadIdx.x & 31;
  const int wave = threadIdx.x >> 5;
  const int task = blockIdx.x * 8 + wave;     // 0..767
  const int mat  = task >> 8;                 // 0..2
  const int n0   = (task & 255) * 16;
  const float* W = (mat == 0) ? Wq : (mat == 1) ? Wk : Wv;
  const int lm   = lane & 15;                 // A row / B col / D col
  const int half = lane >> 4;

  const float* xrow = X + lm * NDIM + 2 * half;   // A: X[m][k + v + 2*half]
  const float* wcol = W + n0 + lm;                // B: W[k + v + 2*half][n0+lm]

  f32x8 acc = {};
  for (int k = 0; k < NDIM; k += 16) {
    if (k + 128 < NDIM)  // stream W ahead (global_prefetch_b8)
      __builtin_prefetch(wcol + (size_t)(k + 128 + 2 * half) * NDIM, 0, 1);
#pragma unroll
    for (int kk = 0; kk < 16; kk += 4) {
      const int kb = k + kk;
      f32x2 a = *(const f32x2*)(xrow + kb);
      f32x2 b;
      b.x = wcol[(size_t)(kb + 2 * half + 0) * NDIM];
      b.y = wcol[(size_t)(kb + 2 * half + 1) * NDIM];
      acc = wmma_f32(a, b, acc);
    }
  }
  float* out = qkv + (size_t)mat * MDIM * NDIM;
#pragma unroll
  for (int v = 0; v < 8; ++v)                  // D: row m = v + 8*half, col lm
    out[(v + 8 * half) * NDIM + n0 + lm] = acc[v];
}

// ---------------- Kernel 2: flash attention per (head, p-split) ----------
__global__ __launch_bounds__(256) void attn_kernel(
    const float* __restrict__ qkv, const float* __restrict__ cacheK,
    const float* __restrict__ cacheV, float* __restrict__ part) {
  __shared__ float q_s[MDIM][DDIM];
  __shared__ float k_s[MDIM][DDIM];
  __shared__ float v_s[MDIM][DDIM];
  __shared__ float w_s[8][16][16];       // per-wave exp(scores)^T staging
  __shared__ float o_sum[MDIM][DDIM];    // block-combined numerator
  __shared__ float red_max[8][16], red_sum[8][16];
  __shared__ float gmax_s[16], gden_s[16];

  const int tid  = threadIdx.x;
  const int lane = tid & 31, wave = tid >> 5;
  const int lm   = lane & 15, half = lane >> 4;
  const int hb = blockIdx.x, h = hb >> 2, s = hb & 3;

  const float* Q = qkv;
  const float* K = qkv + MDIM * NDIM;
  const float* V = qkv + 2 * MDIM * NDIM;

  for (int i = tid; i < MDIM * DDIM; i += 256) {
    const int r = i >> 7, c = i & 127;
    q_s[r][c] = Q[r * NDIM + h * DDIM + c];
    k_s[r][c] = K[r * NDIM + h * DDIM + c];
    v_s[r][c] = V[r * NDIM + h * DDIM + c];
    o_sum[r][c] = 0.0f;
  }
  if (tid < 16) gden_s[tid] = 0.0f;
  __syncthreads();

  // un-affine RMS norm of q and new-k rows (no eps), 32 rows total
  if (tid < 32) {
    const int r = tid & 15;
    float* buf = (tid < 16) ? q_s[r] : k_s[r];
    float ss = 0.0f;
    for (int c = 0; c < DDIM; ++c) ss += buf[c] * buf[c];
    const float sc = rsqrtf(ss * (1.0f / 128.0f));
    for (int c = 0; c < DDIM; ++c) buf[c] *= sc;
  }
  __syncthreads();

  const float* Kc = cacheK + (size_t)h * PDIM * DDIM;
  const float* Vc = cacheV + (size_t)h * PDIM * DDIM;

  float rmax = -3.0e38f, rsum = 0.0f;
  f32x8 oacc[8] = {};                    // out^T tiles: d = dt*16 + v + 8*half, m = lm
  const int t0 = s * TPS;
  const int t1 = (t0 + TPS < NTILES) ? (t0 + TPS) : NTILES;

  for (int t = t0 + wave; t < t1; t += 8) {
    const bool newt = (t == NTILES - 1);   // tile of 16 freshly-projected rows
    // ---- GEMM 1: scores^T[p][m] = sum_d K[p][d] * q[m][d]
    f32x8 sT = {};
#pragma unroll 8
    for (int d = 0; d < DDIM; d += 4) {
      f32x2 a, b;
      if (newt) {
        a.x = k_s[lm][d + 2 * half];
        a.y = k_s[lm][d + 2 * half + 1];
      } else {
        a = *(const f32x2*)(Kc + ((size_t)t * 16 + lm) * DDIM + d + 2 * half);
      }
      b.x = q_s[lm][d + 2 * half];
      b.y = q_s[lm][d + 2 * half + 1];
      sT = wmma_f32(a, b, sT);
    }
    // ---- online softmax (per column m = lm; p spans v and both halves)
    float tmax = sT[0];
#pragma unroll
    for (int v = 1; v < 8; ++v) tmax = fmaxf(tmax, sT[v]);
    tmax = fmaxf(tmax, __shfl_xor(tmax, 16, 32));
    const float nmax = fmaxf(rmax, tmax);
    const float corr = __expf(rmax - nmax);
    rsum *= corr;
#pragma unroll
    for (int i = 0; i < 8; ++i)
#pragma unroll
      for (int v = 0; v < 8; ++v) oacc[i][v] *= corr;
    float psum = 0.0f;
    f32x8 w;
#pragma unroll
    for (int v = 0; v < 8; ++v) {
      const float e = __expf(sT[v] - nmax);
      w[v] = e;
      psum += e;
    }
    psum += __shfl_xor(psum, 16, 32);
    rmax = nmax;
    rsum += psum;
    // stage w^T[p][m] through per-wave LDS (in-order DS, no barrier needed)
#pragma unroll
    for (int v = 0; v < 8; ++v) w_s[wave][v + 8 * half][lm] = w[v];
    // ---- GEMM 2: out^T[d][m] += V^T[d][p] * w^T[p][m]
#pragma unroll
    for (int dt = 0; dt < 8; ++dt) {
      const int d0 = dt * 16;
#pragma unroll
      for (int c = 0; c < 4; ++c) {
        const int p0 = 4 * c + 2 * half;
        f32x2 a, b;
        if (newt) {
          a.x = v_s[p0 + 0][d0 + lm];
          a.y = v_s[p0 + 1][d0 + lm];
        } else {
          const float* vb = Vc + ((size_t)t * 16 + p0) * DDIM + d0 + lm;
          a.x = vb[0];
          a.y = vb[DDIM];
        }
        b.x = w_s[wave][p0 + 0][lm];
        b.y = w_s[wave][p0 + 1][lm];
        oacc[dt] = wmma_f32(a, b, oacc[dt]);
      }
    }
  }

  // ---- merge 8 waves within the block
  if (half == 0) { red_max[wave][lm] = rmax; red_sum[wave][lm] = rsum; }
  __syncthreads();
  if (tid < 16) {
    float g = red_max[0][tid];
    for (int ww = 1; ww < 8; ++ww) g = fmaxf(g, red_max[ww][tid]);
    gmax_s[tid] = g;
  }
  __syncthreads();
  {
    const float corr = __expf(rmax - gmax_s[lm]);
    if (half == 0) atomicAdd(&gden_s[lm], corr * rsum);
#pragma unroll
    for (int dt = 0; dt < 8; ++dt)
#pragma unroll
      for (int v = 0; v < 8; ++v)
        atomicAdd(&o_sum[lm][dt * 16 + v + 8 * half], corr * oacc[dt][v]);
  }
  __syncthreads();

  // ---- emit split partial: [max16 | sum16 | O 16x128]
  float* pb = part + (size_t)hb * PARTF;
  if (tid < 16) { pb[tid] = gmax_s[tid]; pb[16 + tid] = gden_s[tid]; }
  for (int i = tid; i < MDIM * DDIM; i += 256) pb[32 + i] = o_sum[i >> 7][i & 127];
}

// ---------------- Kernel 3: combine the 4 p-splits per head --------------
__global__ __launch_bounds__(256) void combine_kernel(
    const float* __restrict__ part, float* __restrict__ out) {
  __shared__ float gmx[16];
  const int h = blockIdx.x;
  const float* base = part + (size_t)h * SPLITS * PARTF;
  if (threadIdx.x < 16) {
    float g = base[threadIdx.x];
    for (int s = 1; s < SPLITS; ++s) g = fmaxf(g, base[s * PARTF + threadIdx.x]);
    gmx[threadIdx.x] = g;
  }
  __syncthreads();
  for (int i = threadIdx.x; i < MDIM * DDIM; i += 256) {
    const int m = i >> 7, d = i & 127;
    const float g = gmx[m];
    float num = 0.0f, den = 0.0f;
    for (int s = 0; s < SPLITS; ++s) {
      const float* pb = base + s * PARTF;
      const float c = __expf(pb[m] - g);
      num += c * pb[32 + i];
      den += c * pb[16 + m];
    }
    out[m * NDIM + h * DDIM + d] = num / den;
  }
}

extern "C" void kernel_launch(void* const* d_in, const int* in_sizes, int n_in,
                              void* d_out, int out_size, void* d_ws, size_t ws_size,
                              hipStream_t stream) {
  (void)in_sizes; (void)n_in; (void)out_size; (void)ws_size;
  const float* X  = (const float*)d_in[0];
  const float* Wq = (const float*)d_in[1];
  const float* Wk = (const float*)d_in[2];
  const float* Wv = (const float*)d_in[3];
  const float* cK = (const float*)d_in[4];
  const float* cV = (const float*)d_in[5];
  float* out  = (float*)d_out;
  float* qkv  = (float*)d_ws;                      // 3*16*4096 floats
  float* part = qkv + 3 * MDIM * NDIM;             // 128 * 2080 floats

  qkv_kernel<<<96, 256, 0, stream>>>(X, Wq, Wk, Wv, qkv);
  attn_kernel<<<HDIM * SPLITS, 256, 0, stream>>>(qkv, cK, cV, part);
  combine_kernel<<<HDIM, 256, 0, stream>>>(part, out);
}